// UUGCNLayer_7997229105212
// MI455X (gfx1250) — compile-verified
//
#include <hip/hip_runtime.h>
#include <stdint.h>

// GCN aggregation: rst = Din^-1/2 * A^T * Dout^-1/2 * U   (N=100000, E=1200000, D=64)
// Memory-bound (~0.35 GB moved, 0.15 GFLOP => ~15us at 23.3 TB/s): optimize data
// movement. WMMA is deliberately not used: a sparse scatter-add has 0.4 FLOP/byte;
// any dense-matrix formulation wastes MACs and adds traffic. The CDNA5 feature that
// fits this workload is the Tensor Data Mover's gather mode (row-gather DMA to LDS),
// used double-buffered in the aggregation kernel below.

#define D_FEAT 64
#define D2     32   // float2 per row

typedef unsigned int u32;

// ------------------------------------------------------------------
// small utility kernels
// ------------------------------------------------------------------
__global__ void k_zero_u32(u32* __restrict__ p, long n) {
  long stride = (long)gridDim.x * blockDim.x;
  for (long i = (long)blockIdx.x * blockDim.x + threadIdx.x; i < n; i += stride) p[i] = 0u;
}

__global__ void k_zero_f32(float* __restrict__ p, long n) {
  long stride = (long)gridDim.x * blockDim.x;
  for (long i = (long)blockIdx.x * blockDim.x + threadIdx.x; i < n; i += stride) p[i] = 0.0f;
}

// ------------------------------------------------------------------
// degree counting
// ------------------------------------------------------------------
__global__ void k_count_u32(const int* __restrict__ src, const int* __restrict__ dst,
                            u32* __restrict__ oc, u32* __restrict__ ic, int e) {
  int stride = gridDim.x * blockDim.x;
  for (int i = blockIdx.x * blockDim.x + threadIdx.x; i < e; i += stride) {
    atomicAdd(&oc[src[i]], 1u);
    atomicAdd(&ic[dst[i]], 1u);
  }
}

__global__ void k_count_f32(const int* __restrict__ src, const int* __restrict__ dst,
                            float* __restrict__ oc, float* __restrict__ ic, int e) {
  int stride = gridDim.x * blockDim.x;
  for (int i = blockIdx.x * blockDim.x + threadIdx.x; i < e; i += stride) {
    atomicAdd(&oc[src[i]], 1.0f);
    atomicAdd(&ic[dst[i]], 1.0f);
  }
}

__global__ void k_rsqrt_from_u32(const u32* __restrict__ cnt, float* __restrict__ s, int n) {
  int stride = gridDim.x * blockDim.x;
  for (int i = blockIdx.x * blockDim.x + threadIdx.x; i < n; i += stride) {
    u32 c = cnt[i];
    s[i] = rsqrtf((float)(c ? c : 1u));
  }
}

__global__ void k_rsqrt_inplace_f32(float* __restrict__ s, int n) {
  int stride = gridDim.x * blockDim.x;
  for (int i = blockIdx.x * blockDim.x + threadIdx.x; i < n; i += stride)
    s[i] = rsqrtf(fmaxf(s[i], 1.0f));
}

// ------------------------------------------------------------------
// single-block exclusive scan of in-degree -> CSR row offsets
// (98 chunks of 1024; launch-setup cost, negligible vs. gather phase)
// ------------------------------------------------------------------
__global__ void k_scan_block(const u32* __restrict__ cnt, u32* __restrict__ row_start,
                             u32* __restrict__ cursor, int n) {
  __shared__ u32 sh[1024];
  const int t = threadIdx.x;
  u32 carry = 0u;
  for (int base = 0; base < n; base += 1024) {
    int i = base + t;
    u32 v = (i < n) ? cnt[i] : 0u;
    sh[t] = v;
    __syncthreads();
    for (int off = 1; off < 1024; off <<= 1) {
      u32 x = (t >= off) ? sh[t - off] : 0u;
      __syncthreads();
      sh[t] += x;
      __syncthreads();
    }
    u32 incl  = sh[t];
    u32 total = sh[1023];
    if (i < n) {
      u32 ex = carry + incl - v;
      row_start[i] = ex;
      cursor[i]    = ex;
    }
    carry += total;
    __syncthreads();
  }
  if (t == 0) row_start[n] = carry;
}

// scatter edge sources into dst-sorted order (2.4M u32 atomics total vs 77M f32)
__global__ void k_fill(const int* __restrict__ src, const int* __restrict__ dst,
                       u32* __restrict__ cursor, int* __restrict__ src_sorted, int e) {
  int stride = gridDim.x * blockDim.x;
  for (int i = blockIdx.x * blockDim.x + threadIdx.x; i < e; i += stride) {
    u32 pos = atomicAdd(&cursor[dst[i]], 1u);
    src_sorted[pos] = src[i];
  }
}

// ------------------------------------------------------------------
// plain CSR aggregation (VGPR gather path, kept as reference/fallback):
// one wave32 per dst node, lane owns a float2; prefetch hides latency.
// ------------------------------------------------------------------
__global__ void k_aggregate(const float2* __restrict__ uf2, const u32* __restrict__ row_start,
                            const int* __restrict__ src_sorted, const float* __restrict__ s_out,
                            float2* __restrict__ out2, int n) {
  const int lane   = threadIdx.x & 31;
  const int wave   = blockIdx.x * (blockDim.x >> 5) + (threadIdx.x >> 5);
  const int nwaves = gridDim.x * (blockDim.x >> 5);
  for (int node = wave; node < n; node += nwaves) {
    const u32 beg = row_start[node];
    const u32 end = row_start[node + 1];
    float2 acc = make_float2(0.0f, 0.0f);
    for (u32 j = beg; j < end; ++j) {
      if (j + 2 < end)  // hide random-gather latency: global_prefetch_b8
        __builtin_prefetch(&uf2[(size_t)src_sorted[j + 2] * D2 + lane], 0, 1);
      int   s = src_sorted[j];
      float w = s_out[s];
      float2 v = uf2[(size_t)s * D2 + lane];
      acc.x += v.x * w;
      acc.y += v.y * w;
    }
    u32  deg = end - beg;
    float si = rsqrtf((float)(deg ? deg : 1u));
    acc.x *= si;
    acc.y *= si;
    out2[(size_t)node * D2 + lane] = acc;
  }
}

// ------------------------------------------------------------------
// TDM gather-mode aggregation (primary path): D# gather_mode=1, 32-bit
// indices, 8 rows (2KB) per DMA op, DOUBLE-BUFFERED so the TDM engine
// gathers chunk c+1 while the wave accumulates chunk c from LDS.
// s_wait_tensorcnt(1) retires only the older op (same-wave TDM ops
// complete in order). WAR on an LDS buffer is safe: ds_load results are
// consumed by VALU (compiler-inserted s_wait_dscnt) before the next
// tensor_load_to_lds targeting that buffer is issued.
// ------------------------------------------------------------------
typedef __attribute__((ext_vector_type(4))) u32 v4u_;
typedef __attribute__((ext_vector_type(8))) int v8i_;
typedef __attribute__((ext_vector_type(4))) int v4i_;

__device__ __forceinline__ void tdm_gather_rows(u32 lds_off, const float* base, int n_rows,
                                                u32 nvalid, const int* idx /*[8]*/) {
  u32 alo = (u32)__builtin_amdgcn_readfirstlane((int)(uintptr_t)base);
  u32 ahi = (u32)__builtin_amdgcn_readfirstlane((int)((uintptr_t)base >> 32));
  // D# group 0: count=1, gather_index_size=32b, gather_mode=1, lds_addr, global_addr, type=2
  v4u_ g0 = { 1u | (1u << 30) | (1u << 31),
              lds_off,
              alo,
              (ahi & 0x01FFFFFFu) | (2u << 30) };
  // D# group 1: data_size=4B, tensor_dim0=64, tensor_dim1=n_rows, tile_dim0=64,
  //             tile_dim1=#valid indices, tensor_dim0_stride=64
  v8i_ g1 = { (int)(2u << 16),
              (int)(((u32)D_FEAT) << 16),
              (int)(((u32)n_rows & 0xFFFFu) << 16),
              (int)((((u32)n_rows) >> 16) | (((u32)D_FEAT) << 16)),
              (int)nvalid,
              (int)D_FEAT,
              0, 0 };
  v4i_ g2 = { idx[0], idx[1], idx[2], idx[3] };  // row_index_0..3 (32-bit mode)
  v4i_ g3 = { idx[4], idx[5], idx[6], idx[7] };  // row_index_4..7
#if __has_include(<hip/amd_detail/amd_gfx1250_TDM.h>)
  v8i_ gz = { 0, 0, 0, 0, 0, 0, 0, 0 };
  __builtin_amdgcn_tensor_load_to_lds(g0, g1, g2, g3, gz, 0);   // clang-23 / therock-10.0
#else
  __builtin_amdgcn_tensor_load_to_lds(g0, g1, g2, g3, 0);       // ROCm 7.2 / clang-22
#endif
}

__global__ void k_aggregate_tdm(const float* __restrict__ uf, const u32* __restrict__ row_start,
                                const int* __restrict__ src_sorted, const float* __restrict__ s_out,
                                float2* __restrict__ out2, int n) {
  // 8 waves * 2 buffers * 8 rows * 256B = 32 KB (10% of the WGP's 320KB LDS)
  __shared__ float ldsbuf[8 * 2 * 8 * D_FEAT];
  const int waveInBlk = threadIdx.x >> 5;
  const int lane      = threadIdx.x & 31;
  float* buf0 = &ldsbuf[waveInBlk * (2 * 8 * D_FEAT)];
  float* buf1 = buf0 + 8 * D_FEAT;
  const u32 lds_off0 = (u32)__builtin_amdgcn_readfirstlane((int)(uintptr_t)buf0);
  const u32 lds_off1 = (u32)__builtin_amdgcn_readfirstlane((int)(uintptr_t)buf1);
  const int wave     = blockIdx.x * (blockDim.x >> 5) + waveInBlk;
  const int nwaves   = gridDim.x * (blockDim.x >> 5);

  for (int node = wave; node < n; node += nwaves) {
    const u32 beg = row_start[node];
    const u32 end = row_start[node + 1];
    const u32 deg = end - beg;
    float2 acc = make_float2(0.0f, 0.0f);

    if (deg > 0u) {
      const u32 nchunks = (deg + 7u) >> 3;
      int idxbuf[2][8];

      // prime buffer 0 with chunk 0
      {
        const u32 nv0 = (deg < 8u) ? deg : 8u;
#pragma unroll
        for (int k = 0; k < 8; ++k)
          idxbuf[0][k] =
              __builtin_amdgcn_readfirstlane(src_sorted[beg + ((u32)k < nv0 ? (u32)k : 0u)]);
        tdm_gather_rows(lds_off0, uf, n, nv0, idxbuf[0]);
      }

      int cur = 0;
      for (u32 c = 0; c < nchunks; ++c) {
        const u32 j  = beg + (c << 3);
        const u32 nv = (end - j < 8u) ? (end - j) : 8u;

        // issue DMA for chunk c+1 into the other buffer while c is in flight
        if (c + 1u < nchunks) {
          const u32 jn  = j + 8u;
          const u32 nvn = (end - jn < 8u) ? (end - jn) : 8u;
#pragma unroll
          for (int k = 0; k < 8; ++k)
            idxbuf[cur ^ 1][k] =
                __builtin_amdgcn_readfirstlane(src_sorted[jn + ((u32)k < nvn ? (u32)k : 0u)]);
          tdm_gather_rows(cur ? lds_off0 : lds_off1, uf, n, nvn, idxbuf[cur ^ 1]);
          __builtin_amdgcn_s_wait_tensorcnt(1);  // chunk c landed; c+1 still in flight
        } else {
          __builtin_amdgcn_s_wait_tensorcnt(0);  // last chunk landed
        }

        const float* mybuf = cur ? buf1 : buf0;
        for (u32 k = 0; k < nv; ++k) {
          float  w = s_out[idxbuf[cur][k]];
          float2 v = ((const float2*)&mybuf[k * D_FEAT])[lane];
          acc.x += v.x * w;
          acc.y += v.y * w;
        }
        cur ^= 1;
      }
    }

    float si = rsqrtf((float)(deg ? deg : 1u));
    acc.x *= si;
    acc.y *= si;
    out2[(size_t)node * D2 + lane] = acc;
  }
}

// ------------------------------------------------------------------
// fallback path (small workspace): direct atomic scatter-add
// ------------------------------------------------------------------
__global__ void k_scatter_atomic(const float2* __restrict__ uf2, const int* __restrict__ src,
                                 const int* __restrict__ dst, const float* __restrict__ s_out,
                                 float* __restrict__ out, long total) {
  long stride = (long)gridDim.x * blockDim.x;
  for (long t = (long)blockIdx.x * blockDim.x + threadIdx.x; t < total; t += stride) {
    int e    = (int)(t >> 5);
    int lane = (int)(t & 31);
    int s = src[e], d = dst[e];
    float  w = s_out[s];
    float2 v = uf2[(size_t)s * D2 + lane];
    atomicAdd(&out[(size_t)d * D_FEAT + lane * 2 + 0], v.x * w);
    atomicAdd(&out[(size_t)d * D_FEAT + lane * 2 + 1], v.y * w);
  }
}

__global__ void k_scale_rows(float* __restrict__ out, const float* __restrict__ s_in, long total) {
  long stride = (long)gridDim.x * blockDim.x;
  for (long i = (long)blockIdx.x * blockDim.x + threadIdx.x; i < total; i += stride)
    out[i] *= s_in[i >> 6];
}

// ------------------------------------------------------------------
// launch
// ------------------------------------------------------------------
static inline int nblk(long work, int bs, int cap) {
  long b = (work + bs - 1) / bs;
  if (b > cap) b = cap;
  if (b < 1) b = 1;
  return (int)b;
}

extern "C" void kernel_launch(void* const* d_in, const int* in_sizes, int n_in,
                              void* d_out, int out_size, void* d_ws, size_t ws_size,
                              hipStream_t stream) {
  const float* u_f = (const float*)d_in[0];
  const int*   src = (const int*)d_in[1];
  const int*   dst = (const int*)d_in[2];
  float*       out = (float*)d_out;
  const int n = in_sizes[0] / D_FEAT;
  const int e = in_sizes[1];
  const int BS = 256;

  char* ws = (char*)d_ws;
  auto al = [](size_t x) { return (x + 255) & ~(size_t)255; };
  size_t off = 0;
  size_t o_outcnt = off; off = al(off + (size_t)n * 4);
  size_t o_incnt  = off; off = al(off + (size_t)n * 4);
  size_t o_sout   = off; off = al(off + (size_t)n * 4);
  size_t o_rs     = off; off = al(off + (size_t)(n + 1) * 4);
  size_t o_cur    = off; off = al(off + (size_t)n * 4);
  size_t o_srt    = off; off = al(off + (size_t)e * 4);
  const size_t need_csr = off;

  if (ws_size >= need_csr) {
    // ---- CSR path: build dst-sorted edge list, aggregate without atomics ----
    u32*   out_cnt    = (u32*)(ws + o_outcnt);
    u32*   in_cnt     = (u32*)(ws + o_incnt);
    float* s_out      = (float*)(ws + o_sout);
    u32*   row_start  = (u32*)(ws + o_rs);
    u32*   cursor     = (u32*)(ws + o_cur);
    int*   src_sorted = (int*)(ws + o_srt);

    k_zero_u32<<<nblk(n, BS, 8192), BS, 0, stream>>>(out_cnt, n);
    k_zero_u32<<<nblk(n, BS, 8192), BS, 0, stream>>>(in_cnt, n);
    k_count_u32<<<nblk(e, BS, 8192), BS, 0, stream>>>(src, dst, out_cnt, in_cnt, e);
    k_rsqrt_from_u32<<<nblk(n, BS, 8192), BS, 0, stream>>>(out_cnt, s_out, n);
    k_scan_block<<<1, 1024, 0, stream>>>(in_cnt, row_start, cursor, n);
    k_fill<<<nblk(e, BS, 8192), BS, 0, stream>>>(src, dst, cursor, src_sorted, e);

    const bool use_tdm = true;  // tensor_load_to_lds lowering confirmed by histogram
    if (use_tdm) {
      k_aggregate_tdm<<<nblk((long)n * 32, BS, 16384), BS, 0, stream>>>(
          u_f, row_start, src_sorted, s_out, (float2*)out, n);
    } else {
      k_aggregate<<<nblk((long)n * 32, BS, 16384), BS, 0, stream>>>(
          (const float2*)u_f, row_start, src_sorted, s_out, (float2*)out, n);
    }
  } else {
    // ---- fallback: atomic scatter-add (needs only 2N floats of scratch) ----
    float* s_out = (float*)(ws + 0);
    float* s_in  = (float*)(ws + al((size_t)n * 4));

    k_zero_f32<<<nblk(n, BS, 8192), BS, 0, stream>>>(s_out, n);
    k_zero_f32<<<nblk(n, BS, 8192), BS, 0, stream>>>(s_in, n);
    k_count_f32<<<nblk(e, BS, 8192), BS, 0, stream>>>(src, dst, s_out, s_in, e);
    k_rsqrt_inplace_f32<<<nblk(n, BS, 8192), BS, 0, stream>>>(s_out, n);
    k_rsqrt_inplace_f32<<<nblk(n, BS, 8192), BS, 0, stream>>>(s_in, n);
    k_zero_f32<<<nblk((long)n * D_FEAT, BS, 16384), BS, 0, stream>>>(out, (long)n * D_FEAT);
    k_scatter_atomic<<<nblk((long)e * 32, BS, 16384), BS, 0, stream>>>(
        (const float2*)u_f, src, dst, s_out, out, (long)e * 32);
    k_scale_rows<<<nblk((long)n * D_FEAT, BS, 16384), BS, 0, stream>>>(out, s_in, (long)n * D_FEAT);
  }
}